// SampleAndAggregate_51419348467782
// MI455X (gfx1250) — compile-verified
//
#include <hip/hip_runtime.h>
#include <math.h>

// ---------------------------------------------------------------------------
// GraphSAGE (2-hop mean aggregator) + VAE head for MI455X (gfx1250).
// - All matmuls on the WMMA pipe: V_WMMA_F32_16X16X4_F32 (exact f32 numerics).
// - Weight tiles staged into LDS *transposed* via the gfx1250 async-to-LDS
//   DMA path (global_load_async_to_lds_b32 + s_wait_asynccnt), bank-pad K+4.
// - Binding resource is the 131 MB hop-2 feature gather (L2-resident: the
//   51 MB feature table fits in the 192 MB L2); compute is ~1.5 GFLOP.
// ---------------------------------------------------------------------------

typedef __attribute__((ext_vector_type(2))) float v2f;
typedef __attribute__((ext_vector_type(8))) float v8f;

#define N_NODES 100000
#define FDIM    128
#define DDEG    64
#define BATCH   1024
#define S1N     10
#define S2N     25
#define HDIM    256

#define ACT_NONE 0
#define ACT_RELU 1
#define ACT_SIG  2

// ---- index sampling -------------------------------------------------------
__global__ void idx1_kernel(const int* __restrict__ adj, const int* __restrict__ batch1,
                            const int* __restrict__ cols1, int* __restrict__ s1) {
  int i = blockIdx.x * blockDim.x + threadIdx.x;
  if (i >= BATCH * S1N) return;
  int b = i / S1N;
  s1[i] = adj[(size_t)batch1[b] * DDEG + cols1[i]];
}

__global__ void idx2_kernel(const int* __restrict__ adj, const int* __restrict__ s1,
                            const int* __restrict__ cols2, int* __restrict__ s2) {
  int i = blockIdx.x * blockDim.x + threadIdx.x;
  if (i >= BATCH * S1N * S2N) return;
  int r = i / S2N;
  s2[i] = adj[(size_t)s1[r] * DDEG + cols2[i]];
}

// ---- gathered row mean: out[p][f] = mean_q src[idx[p*S+q]][f] -------------
// idx == nullptr -> contiguous rows (row = p*S+q). blockDim.x == F.
__global__ void mean_rows_kernel(const float* __restrict__ src, const int* __restrict__ idx,
                                 int S, int F, float* __restrict__ out) {
  int p = blockIdx.x;
  int f = threadIdx.x;
  float acc = 0.0f;
  for (int q = 0; q < S; ++q) {
    int row = idx ? idx[p * S + q] : (p * S + q);
    acc += src[(size_t)row * F + f];
  }
  out[(size_t)p * F + f] = acc * (1.0f / (float)S);
}

// ---- WMMA f32 GEMM --------------------------------------------------------
// C[m][n] = act( sum_k A[gather(m)][k] * W[k][n] + bias[n] )
// block = (32,4): 4 waves, each computes one 16x16 tile (same N-tile,
// consecutive M-tiles). The block's W[:, col0:col0+16] tile is DMA'd into
// LDS transposed (wtile[c][k]) with the gfx1250 async-to-LDS path, pad K+4
// so the half-wave b64 reads in the WMMA loop are bank-conflict free.
//
// f32 WMMA operand layout (ISA 7.12.2): A 16x4 -> lanes 0-15 hold (K=k,k+1)
// of row M=lane, lanes 16-31 hold (K=k+2,k+3) of row M=lane-16; B mirrored;
// C/D: VGPR v holds M=v (lanes 0-15) / M=v+8 (lanes 16-31), N=lane&15.
template<int K, int ACT>
__global__ __launch_bounds__(128)
void gemm_wmma_f32(const float* __restrict__ A, const int* __restrict__ gather,
                   int lda, const float* __restrict__ W, int ldw,
                   const float* __restrict__ bias,
                   float* __restrict__ C, int ldc) {
  constexpr int WPAD = K + 4;                // bank-conflict-free LDS stride
  __shared__ float wtile[16 * WPAD];

  const int lane = threadIdx.x;              // 0..31
  const int wave = threadIdx.y;              // 0..3
  const int tid  = wave * 32 + lane;
  const int col0 = blockIdx.x * 16;

  // ---- async DMA transpose-stage: wtile[c*WPAD + k] = W[k][col0 + c] -----
  for (int e = tid; e < K * 16; e += 128) {  // uniform trip count, EXEC full
    int k = e >> 4;
    int c = e & 15;
    unsigned lds_off = (unsigned)(uintptr_t)&wtile[c * WPAD + k];  // addr[31:0]
    unsigned long long gaddr =
        (unsigned long long)(uintptr_t)(W + (size_t)k * ldw + (col0 + c));
    asm volatile("global_load_async_to_lds_b32 %0, %1, off"
                 :: "v"(lds_off), "v"(gaddr)
                 : "memory");
  }
  asm volatile("s_wait_asynccnt 0x0" ::: "memory");
  __syncthreads();

  const int mt    = blockIdx.y * 4 + wave;
  const int row0  = mt * 16;
  const int rsub  = lane & 15;
  const int khalf = (lane >> 4) * 2;         // 0 for lanes 0-15, 2 for 16-31

  const int arow = row0 + rsub;
  const int srow = gather ? gather[arow] : arow;
  const float* __restrict__ Arow = A + (size_t)srow * lda + khalf;
  const float* __restrict__ Brow = &wtile[rsub * WPAD + khalf];

  v8f acc = {};
#pragma unroll 8
  for (int k = 0; k < K; k += 4) {
    float2 av = *(const float2*)(Arow + k);  // 8B aligned: k%4==0, khalf even
    float2 bv = *(const float2*)(Brow + k);  // ds_load_b64, conflict-free
    v2f a = {av.x, av.y};
    v2f b = {bv.x, bv.y};
    acc = __builtin_amdgcn_wmma_f32_16x16x4_f32(
        /*neg_a=*/false, a, /*neg_b=*/false, b,
        /*c_mod=*/(short)0, acc, /*reuse_a=*/false, /*reuse_b=*/false);
  }

  const float badd  = bias ? bias[col0 + rsub] : 0.0f;
  const int   rbase = row0 + (lane >> 4) * 8;
#pragma unroll
  for (int v = 0; v < 8; ++v) {
    float x = acc[v] + badd;
    if (ACT == ACT_RELU) x = fmaxf(x, 0.0f);
    if (ACT == ACT_SIG)  x = 1.0f / (1.0f + expf(-x));
    C[(size_t)(rbase + v) * ldc + col0 + rsub] = x;
  }
}

// ---- row-wise L2 normalize (in place), blockDim = HDIM --------------------
__global__ void l2norm_kernel(float* __restrict__ hid) {
  __shared__ float red[HDIM];
  int r = blockIdx.x, t = threadIdx.x;
  float v = hid[(size_t)r * HDIM + t];
  red[t] = v * v;
  __syncthreads();
  for (int s = HDIM / 2; s > 0; s >>= 1) {
    if (t < s) red[t] += red[t + s];
    __syncthreads();
  }
  float nrm = fmaxf(sqrtf(red[0]), 1e-12f);
  hid[(size_t)r * HDIM + t] = v / nrm;
}

// ---- reparameterization: z = mu + exp(0.5*ls)*eps -------------------------
__global__ void reparam_kernel(const float* __restrict__ mu, const float* __restrict__ ls,
                               const float* __restrict__ eps, float* __restrict__ z) {
  int i = blockIdx.x * blockDim.x + threadIdx.x;
  if (i < BATCH * HDIM) z[i] = mu[i] + expf(0.5f * ls[i]) * eps[i];
}

// ---------------------------------------------------------------------------
extern "C" void kernel_launch(void* const* d_in, const int* in_sizes, int n_in,
                              void* d_out, int out_size, void* d_ws, size_t ws_size,
                              hipStream_t stream) {
  (void)in_sizes; (void)n_in; (void)out_size; (void)ws_size;

  const float* features = (const float*)d_in[0];
  const int*   adj      = (const int*)d_in[1];
  const int*   batch1   = (const int*)d_in[2];
  const int*   cols1    = (const int*)d_in[3];
  const int*   cols2    = (const int*)d_in[4];
  const float* W_self0  = (const float*)d_in[5];
  const float* W_neigh0 = (const float*)d_in[6];
  const float* W_self1  = (const float*)d_in[7];
  const float* W_neigh1 = (const float*)d_in[8];
  const float* W_h  = (const float*)d_in[9];   const float* b_h  = (const float*)d_in[10];
  const float* W_mu = (const float*)d_in[11];  const float* b_mu = (const float*)d_in[12];
  const float* W_ls = (const float*)d_in[13];  const float* b_ls = (const float*)d_in[14];
  const float* W_d1 = (const float*)d_in[15];  const float* b_d1 = (const float*)d_in[16];
  const float* W_d2 = (const float*)d_in[17];  const float* b_d2 = (const float*)d_in[18];
  const float* eps  = (const float*)d_in[19];
  float* y = (float*)d_out;

  // workspace carve (~26 MB)
  char* wsp = (char*)d_ws;
  auto carve = [&wsp](size_t bytes) {
    char* p = wsp;
    wsp += (bytes + 255) & ~(size_t)255;
    return p;
  };
  int*   s1  = (int*)carve((size_t)BATCH * S1N * sizeof(int));
  int*   s2  = (int*)carve((size_t)BATCH * S1N * S2N * sizeof(int));
  float* m2  = (float*)carve((size_t)BATCH * S1N * FDIM * sizeof(float));
  float* m1  = (float*)carve((size_t)BATCH * FDIM * sizeof(float));
  float* g1  = (float*)carve((size_t)BATCH * S1N * HDIM * sizeof(float));
  float* g0  = (float*)carve((size_t)BATCH * HDIM * sizeof(float));
  float* mg1 = (float*)carve((size_t)BATCH * HDIM * sizeof(float));
  float* hid = (float*)carve((size_t)BATCH * HDIM * sizeof(float));
  float* he  = (float*)carve((size_t)BATCH * HDIM * sizeof(float));
  float* mu  = (float*)carve((size_t)BATCH * HDIM * sizeof(float));
  float* ls  = (float*)carve((size_t)BATCH * HDIM * sizeof(float));
  float* z   = (float*)carve((size_t)BATCH * HDIM * sizeof(float));
  float* hd  = (float*)carve((size_t)BATCH * HDIM * sizeof(float));

  // --- neighbor sampling ---
  idx1_kernel<<<(BATCH * S1N + 255) / 256, 256, 0, stream>>>(adj, batch1, cols1, s1);
  idx2_kernel<<<(BATCH * S1N * S2N + 255) / 256, 256, 0, stream>>>(adj, s1, cols2, s2);

  // --- neighbor means (fused gather + mean; L2-resident feature table) ---
  mean_rows_kernel<<<BATCH * S1N, FDIM, 0, stream>>>(features, s2, S2N, FDIM, m2);
  mean_rows_kernel<<<BATCH,       FDIM, 0, stream>>>(features, s1, S1N, FDIM, m1);

  dim3 blk(32, 4);
  auto grd = [](int M, int Nout) { return dim3(Nout / 16, M / 64); };

  // --- layer 0: concat(relu(self@Ws), relu(mean@Wn)), gathers fused into A ---
  gemm_wmma_f32<FDIM, ACT_RELU><<<grd(BATCH * S1N, 128), blk, 0, stream>>>(
      features, s1, FDIM, W_self0, 128, nullptr, g1, HDIM);
  gemm_wmma_f32<FDIM, ACT_RELU><<<grd(BATCH * S1N, 128), blk, 0, stream>>>(
      m2, nullptr, FDIM, W_neigh0, 128, nullptr, g1 + 128, HDIM);
  gemm_wmma_f32<FDIM, ACT_RELU><<<grd(BATCH, 128), blk, 0, stream>>>(
      features, batch1, FDIM, W_self0, 128, nullptr, g0, HDIM);
  gemm_wmma_f32<FDIM, ACT_RELU><<<grd(BATCH, 128), blk, 0, stream>>>(
      m1, nullptr, FDIM, W_neigh0, 128, nullptr, g0 + 128, HDIM);

  // --- layer 1 ---
  mean_rows_kernel<<<BATCH, HDIM, 0, stream>>>(g1, nullptr, S1N, HDIM, mg1);
  gemm_wmma_f32<HDIM, ACT_NONE><<<grd(BATCH, 128), blk, 0, stream>>>(
      g0, nullptr, HDIM, W_self1, 128, nullptr, hid, HDIM);
  gemm_wmma_f32<HDIM, ACT_NONE><<<grd(BATCH, 128), blk, 0, stream>>>(
      mg1, nullptr, HDIM, W_neigh1, 128, nullptr, hid + 128, HDIM);
  l2norm_kernel<<<BATCH, HDIM, 0, stream>>>(hid);

  // --- VAE encoder / decoder ---
  gemm_wmma_f32<HDIM, ACT_RELU><<<grd(BATCH, HDIM), blk, 0, stream>>>(
      hid, nullptr, HDIM, W_h, HDIM, b_h, he, HDIM);
  gemm_wmma_f32<HDIM, ACT_NONE><<<grd(BATCH, HDIM), blk, 0, stream>>>(
      he, nullptr, HDIM, W_mu, HDIM, b_mu, mu, HDIM);
  gemm_wmma_f32<HDIM, ACT_NONE><<<grd(BATCH, HDIM), blk, 0, stream>>>(
      he, nullptr, HDIM, W_ls, HDIM, b_ls, ls, HDIM);
  reparam_kernel<<<(BATCH * HDIM + 255) / 256, 256, 0, stream>>>(mu, ls, eps, z);
  gemm_wmma_f32<HDIM, ACT_RELU><<<grd(BATCH, HDIM), blk, 0, stream>>>(
      z, nullptr, HDIM, W_d1, HDIM, b_d1, hd, HDIM);
  gemm_wmma_f32<HDIM, ACT_SIG><<<grd(BATCH, HDIM), blk, 0, stream>>>(
      hd, nullptr, HDIM, W_d2, HDIM, b_d2, y, HDIM);
}